// InputLayer_19361712571163
// MI455X (gfx1250) — compile-verified
//
#include <hip/hip_runtime.h>
#include <math.h>

// ---------------------------------------------------------------------------
// Diffractive input layer (Rayleigh-Sommerfeld) for MI455X / gfx1250.
//
// Algorithmic restructure: identical 96-pitch grids on both planes make the
// 9216x9216 complex GEMV a 2D convolution with a 191x191 complex kernel
// (2300x fewer transcendentals). The convolution is then expressed as
// Out = sum_{dI} Xshift(dI) * Toeplitz(dI) and run on V_WMMA_F32_16X16X4_F32
// (fp32 in/out, matching the reference's complex64 math).
// ---------------------------------------------------------------------------

typedef __attribute__((ext_vector_type(2))) float v2f;
typedef __attribute__((ext_vector_type(8))) float v8f;

#define N_SIDE     96
#define N_ELEMS    (N_SIDE * N_SIDE)       // 9216
#define KSIDE      191                     // 2*96-1 kernel taps per dim
#define KSTRIDE    192                     // padded row stride (floats)
#define KER_ELEMS  (KSIDE * KSTRIDE)       // 36672 floats per (re|im) table
#define WS_KOFF    64                      // ws[0]=invmax; tables at +64

// ---- K0: max-reduce over x (9216 elems), write 1/max to ws[0] -------------
__global__ __launch_bounds__(256) void k_max_reduce(const float* __restrict__ x,
                                                    float* __restrict__ ws) {
    __shared__ float sm[256];
    float m = 0.0f;
    for (int i = threadIdx.x; i < N_ELEMS; i += 256) m = fmaxf(m, x[i]);
    sm[threadIdx.x] = m;
    __syncthreads();
    for (int s = 128; s > 0; s >>= 1) {
        if (threadIdx.x < s) sm[threadIdx.x] = fmaxf(sm[threadIdx.x], sm[threadIdx.x + s]);
        __syncthreads();
    }
    if (threadIdx.x == 0) ws[0] = 1.0f / sm[0];
}

// ---- K1: build 191x191 complex RS kernel table in workspace ---------------
// w(di,dj) = (dz/r^2) * (1/(2*pi*r) - j/lambda) * exp(j*2*pi*r/lambda)
__global__ __launch_bounds__(256) void k_build_table(float* __restrict__ ws) {
    int t = blockIdx.x * blockDim.x + threadIdx.x;
    if (t >= KSIDE * KSIDE) return;
    int ki = t / KSIDE;            // di - si + 95
    int kj = t - ki * KSIDE;       // dj - sj + 95
    const float dl    = 1.0e-3f / 96.0f;
    const float dz    = 1.0e-2f;
    const float lam   = 1.55e-6f;
    const float twopi = 6.28318530717958647692f;
    float ddx = (float)(ki - 95) * dl;
    float ddy = (float)(kj - 95) * dl;
    float r2  = ddx * ddx + ddy * ddy + dz * dz;
    float r   = sqrtf(r2);
    float are =  dz / (r2 * twopi * r);   // Re{(dz/r^2)(1/(2 pi r) + 1/(j lam))}
    float aim = -dz / (r2 * lam);         // Im{...} = -dz/(r^2 lam)
    float phase = (twopi / lam) * r;      // fp32, like the reference
    float sn, cs;
    sincosf(phase, &sn, &cs);
    float* Kre = ws + WS_KOFF;
    float* Kim = Kre + KER_ELEMS;
    Kre[ki * KSTRIDE + kj] = are * cs - aim * sn;
    Kim[ki * KSTRIDE + kj] = are * sn + aim * cs;
}

// ---- K2: convolution as sum of Toeplitz GEMMs on V_WMMA_F32_16X16X4_F32 ---
// One wave32 per 16x16 output tile (6x6 tiles). For each row-shift dI and
// each 4-wide sj chunk:
//   A[m][k] = x[mb*16+m-dI][sjc*4+k]            (zero outside [0,96))
//   B[k][n] = Ker[dI+95][(nb*16+n)-(sjc*4+k)+95]
// Accumulate Re and Im tiles; scale by 1/max at store (linearity).
__global__ __launch_bounds__(32) void k_conv_wmma(const float* __restrict__ x,
                                                  const float* __restrict__ ws,
                                                  float* __restrict__ out) {
    const float* __restrict__ Kre = ws + WS_KOFF;
    const float* __restrict__ Kim = Kre + KER_ELEMS;
    const float invmax = ws[0];

    const int mb = blockIdx.x / 6;        // output row-tile (di / 16)
    const int nb = blockIdx.x % 6;        // output col-tile (dj / 16)
    const int l  = threadIdx.x;           // wave32 lane
    const int lo = l & 15;
    const int hi = l >> 4;
    const int dj = nb * 16 + lo;          // this lane's N index

    v8f cre = {};
    v8f cim = {};

    // dI = di - si; valid range for this 16-row tile: [mb*16-95, mb*16+15]
    const int dImin = mb * 16 - 95;
    const int dImax = mb * 16 + 15;
    for (int dI = dImin; dI <= dImax; ++dI) {
        const int ki = dI + 95;                                   // 0..190
        const float* kr = Kre + ki * KSTRIDE + 95 + dj;           // index -sj
        const float* kq = Kim + ki * KSTRIDE + 95 + dj;
        __builtin_prefetch(kr + KSTRIDE, 0, 0);                   // next row

        const int  sr    = mb * 16 + lo - dI;                     // si for A row lo
        const bool valid = (sr >= 0) & (sr < N_SIDE);
        const float* xrow = x + (valid ? sr : 0) * N_SIDE;        // safe addr

        for (int sjc = 0; sjc < 24; ++sjc) {
            const int sj0 = sjc * 4 + 2 * hi;                     // K lanes {0,1}|{2,3}
            v2f a;
            a.x = valid ? xrow[sj0]     : 0.0f;
            a.y = valid ? xrow[sj0 + 1] : 0.0f;
            // B[k][n]: dj - sj + 95 always in [0,190] -> no bounds check
            v2f bre, bim;
            bre.x = kr[-sj0];
            bre.y = kr[-sj0 - 1];
            bim.x = kq[-sj0];
            bim.y = kq[-sj0 - 1];
            cre = __builtin_amdgcn_wmma_f32_16x16x4_f32(
                      false, a, false, bre, (short)0, cre, false, false);
            cim = __builtin_amdgcn_wmma_f32_16x16x4_f32(
                      false, a, false, bim, (short)0, cim, false, false);
        }
    }

    // C/D layout: VGPR g holds element [M = g + 8*hi, N = lo].
    // Output is complex64-compatible: interleaved (re, im) float pairs.
    float2* __restrict__ o2 = (float2*)out;
#pragma unroll
    for (int g = 0; g < 8; ++g) {
        const int di = mb * 16 + g + 8 * hi;
        o2[di * N_SIDE + dj] = make_float2(cre[g] * invmax, cim[g] * invmax);
    }
}

// ---------------------------------------------------------------------------
extern "C" void kernel_launch(void* const* d_in, const int* in_sizes, int n_in,
                              void* d_out, int out_size, void* d_ws, size_t ws_size,
                              hipStream_t stream) {
    (void)in_sizes; (void)n_in; (void)out_size; (void)ws_size;
    const float* x  = (const float*)d_in[0];
    float*       ws = (float*)d_ws;          // [0]=invmax, +64: Kre, Kim tables
    float*       out = (float*)d_out;        // 9216 complex64 = 18432 floats

    k_max_reduce<<<1, 256, 0, stream>>>(x, ws);
    k_build_table<<<(KSIDE * KSIDE + 255) / 256, 256, 0, stream>>>(ws);
    k_conv_wmma<<<36, 32, 0, stream>>>(x, ws, out);
}